// WCSS_49649821941971
// MI455X (gfx1250) — compile-verified
//
#include <hip/hip_runtime.h>
#include <hip/hip_bf16.h>

typedef __attribute__((ext_vector_type(2))) float v2f;
typedef __attribute__((ext_vector_type(8))) float v8f;

#define N_SAMPLES 8192
#define D_ELEMS   6144          // C*T = 3*2048
#define K_CLASS   10
#define KPAD      16
#define CHUNK     256           // samples per block (y dim of grid)
#define NCHUNK    (N_SAMPLES / CHUNK)     // 32
#define STRIPS    (D_ELEMS / 16)          // 384 strips of 16 d-values
#define WAVES_PER_BLOCK 4

// ---------------------------------------------------------------------------
// Zero the workspace accumulators (harness poisons ws with 0xAA).
// ---------------------------------------------------------------------------
__global__ void wcss_zero(float* __restrict__ ws, int n) {
    int i = blockIdx.x * blockDim.x + threadIdx.x;
    if (i < n) ws[i] = 0.0f;
}

// ---------------------------------------------------------------------------
// Main single-pass kernel: S[k,d] += onehot * x ; Q[k,d] += onehot * x^2
// via V_WMMA_F32_16X16X4_F32.  A = one-hot (16 classes x 4 samples),
// B = X tile (4 samples x 16 d).  One wave owns one 16-wide d-strip.
// ---------------------------------------------------------------------------
__global__ __launch_bounds__(128) void wcss_wmma(
    const float* __restrict__ X,
    const int*   __restrict__ y2,     // int64 labels viewed as int pairs (LE)
    float* __restrict__ Ssum,         // [KPAD][D_ELEMS]
    float* __restrict__ Qsum)         // [KPAD][D_ELEMS]
{
    __shared__ int ycls[CHUNK];

    const int tid    = threadIdx.x;
    const int lane   = tid & 31;
    const int wave   = tid >> 5;
    const int chunk0 = blockIdx.y * CHUNK;

    // Cooperative load of this chunk's labels (low dword of each int64).
    for (int i = tid; i < CHUNK; i += 128) ycls[i] = y2[2 * (chunk0 + i)];
    __syncthreads();

    const int strip = blockIdx.x * WAVES_PER_BLOCK + wave;   // 0..383
    const int d0    = strip * 16;
    const int n     = lane & 15;                // B col (d offset) == A row (class)
    const int kb0   = (lane < 16) ? 0 : 2;      // which K-pair this half-wave holds

    v8f accS0 = {}, accS1 = {};   // S-tile accumulators (ping-pong)
    v8f accQ0 = {}, accQ1 = {};   // Q-tile accumulators (ping-pong)

    const float* xb = X + (size_t)(chunk0 + kb0) * D_ELEMS + d0 + n;

    #pragma unroll 2
    for (int g = 0; g < CHUNK / 4; ++g) {
        const int i0 = g * 4;
        const int c0 = ycls[i0 + kb0];
        const int c1 = ycls[i0 + kb0 + 1];

        v2f a, b, q;
        a.x = (c0 == n) ? 1.0f : 0.0f;          // one-hot A tile (16x4 f32)
        a.y = (c1 == n) ? 1.0f : 0.0f;

        const float* p = xb + (size_t)i0 * D_ELEMS;
        b.x = p[0];                              // B tile (4x16 f32), coalesced
        b.y = p[D_ELEMS];
        q.x = b.x * b.x;
        q.y = b.y * b.y;

        if (i0 + 8 < CHUNK)                      // wave-uniform: EXEC stays full
            __builtin_prefetch(p + (size_t)8 * D_ELEMS, 0, 0);

        if (g & 1) {
            accS1 = __builtin_amdgcn_wmma_f32_16x16x4_f32(
                        false, a, false, b, (short)0, accS1, false, false);
            accQ1 = __builtin_amdgcn_wmma_f32_16x16x4_f32(
                        false, a, false, q, (short)0, accQ1, false, false);
        } else {
            accS0 = __builtin_amdgcn_wmma_f32_16x16x4_f32(
                        false, a, false, b, (short)0, accS0, false, false);
            accQ0 = __builtin_amdgcn_wmma_f32_16x16x4_f32(
                        false, a, false, q, (short)0, accQ0, false, false);
        }
    }

    // C/D layout: VGPR j = row (j + 8*(lane>=16)), col = lane&15.
    const int rbase = (lane < 16) ? 0 : 8;
    #pragma unroll
    for (int j = 0; j < 8; ++j) {
        const int off = (rbase + j) * D_ELEMS + d0 + n;
        atomicAdd(&Ssum[off], accS0[j] + accS1[j]);
        atomicAdd(&Qsum[off], accQ0[j] + accQ1[j]);
    }
}

// ---------------------------------------------------------------------------
// Per-class reduction over d:  Ak[k] = sum_d S[k,d]^2 ; Bk[k] = sum_d Q[k,d]
// ---------------------------------------------------------------------------
__global__ __launch_bounds__(256) void wcss_reduce(
    const float* __restrict__ Ssum, const float* __restrict__ Qsum,
    float* __restrict__ Ak, float* __restrict__ Bk)
{
    __shared__ float sA[KPAD], sB[KPAD];
    const int t = threadIdx.x;
    if (t < KPAD) { sA[t] = 0.0f; sB[t] = 0.0f; }
    __syncthreads();

    const int idx = blockIdx.x * 256 + t;        // exactly KPAD*D_ELEMS threads
    const int k   = idx / D_ELEMS;
    const float s = Ssum[idx];
    atomicAdd(&sA[k], s * s);
    atomicAdd(&sB[k], Qsum[idx]);
    __syncthreads();

    if (t < KPAD) {
        atomicAdd(&Ak[t], sA[t]);
        atomicAdd(&Bk[t], sB[t]);
    }
}

// ---------------------------------------------------------------------------
// Final: counts from labels, then scalar loss.
// loss = mean_k ( Bk - Ak/n_k ) / (n_k * D)
// ---------------------------------------------------------------------------
__global__ __launch_bounds__(256) void wcss_final(
    const int* __restrict__ y2,
    const float* __restrict__ Ak, const float* __restrict__ Bk,
    float* __restrict__ out)
{
    __shared__ int cnt[KPAD];
    const int t = threadIdx.x;
    if (t < KPAD) cnt[t] = 0;
    __syncthreads();
    for (int i = t; i < N_SAMPLES; i += 256)
        atomicAdd(&cnt[y2[2 * i]], 1);
    __syncthreads();

    if (t == 0) {
        float loss = 0.0f;
        for (int k = 0; k < K_CLASS; ++k) {
            const float nk = (float)cnt[k];
            loss += (Bk[k] - Ak[k] / nk) / (nk * (float)D_ELEMS);
        }
        out[0] = loss / (float)K_CLASS;
    }
}

// ---------------------------------------------------------------------------
extern "C" void kernel_launch(void* const* d_in, const int* in_sizes, int n_in,
                              void* d_out, int out_size, void* d_ws, size_t ws_size,
                              hipStream_t stream) {
    (void)in_sizes; (void)n_in; (void)out_size; (void)ws_size;

    const float* X  = (const float*)d_in[0];
    const int*   y2 = (const int*)d_in[1];   // int64 labels, read low dwords
    float* out = (float*)d_out;

    float* ws   = (float*)d_ws;
    float* Ssum = ws;                         // KPAD * D_ELEMS
    float* Qsum = Ssum + KPAD * D_ELEMS;      // KPAD * D_ELEMS
    float* Ak   = Qsum + KPAD * D_ELEMS;      // KPAD
    float* Bk   = Ak + KPAD;                  // KPAD

    const int ztotal = 2 * KPAD * D_ELEMS + 2 * KPAD;
    wcss_zero<<<(ztotal + 255) / 256, 256, 0, stream>>>(ws, ztotal);

    dim3 grid(STRIPS / WAVES_PER_BLOCK, NCHUNK);   // 96 x 32 blocks, 128 thr
    wcss_wmma<<<grid, 128, 0, stream>>>(X, y2, Ssum, Qsum);

    wcss_reduce<<<(KPAD * D_ELEMS) / 256, 256, 0, stream>>>(Ssum, Qsum, Ak, Bk);
    wcss_final<<<1, 256, 0, stream>>>(y2, Ak, Bk, out);
}